// GAT_44220983280303
// MI455X (gfx1250) — compile-verified
//
#include <hip/hip_runtime.h>
#include <math.h>
#include <stdint.h>

// ---------------- problem constants (from reference setup_inputs) -----------
#define N_NODES 50000
#define E_EDGES 800000
#define ET      (E_EDGES + N_NODES)   // edges + self loops = 850000
#define IN_CH   128
#define HID     32
#define H1      8
#define C1      (H1 * HID)            // 256
#define OUT_CH  64
#define NEG_SLOPE 0.2f
#define EPSF    1e-16f

typedef float v2f __attribute__((ext_vector_type(2)));
typedef float v8f __attribute__((ext_vector_type(8)));
typedef unsigned int u32x4 __attribute__((ext_vector_type(4)));
typedef int i32x4 __attribute__((ext_vector_type(4)));
typedef int i32x8 __attribute__((ext_vector_type(8)));

#if defined(__has_builtin)
# if __has_builtin(__builtin_amdgcn_tensor_load_to_lds)
#  define HAVE_TDM 1
# endif
#endif
#ifndef HAVE_TDM
# define HAVE_TDM 0
#endif

// Compile-probe markers (visible in hipcc stderr): device pass reports whether
// the TDM builtin path is active and which arity variant was selected.
#if HAVE_TDM
# if defined(__clang_major__) && (__clang_major__ >= 23)
#  pragma message("CDNA5 probe: tensor_load_to_lds ENABLED (6-arg, clang>=23)")
# else
#  pragma message("CDNA5 probe: tensor_load_to_lds ENABLED (5-arg, clang<=22)")
# endif
#else
# pragma message("CDNA5 probe: tensor_load_to_lds NOT available; cooperative-copy fallback")
#endif

// ---- order-preserving float <-> uint mapping for atomicMax on floats ------
__device__ __forceinline__ unsigned mapf(float f) {
  unsigned u = __float_as_uint(f);
  return (u & 0x80000000u) ? ~u : (u | 0x80000000u);
}
__device__ __forceinline__ float unmapf(unsigned m) {
  return __uint_as_float((m & 0x80000000u) ? (m ^ 0x80000000u) : ~m);
}

#if HAVE_TDM
// ---------------------------------------------------------------------------
// TDM: DMA a K x W fp32 tile (global row stride = Ncol floats) into LDS,
// packed as contiguous W-float rows. 2-D descriptor per CDNA5 ISA:
//  g0: count=1 | lds_addr | global_addr[56:0] | type=2
//  g1: data_size=4B, tensor_dim0=Ncol, tensor_dim1=K, tile_dim0=W,
//      tile_dim1=K, tensor_dim0_stride=Ncol
//  groups 2/3 zero (tile_dim2=0 -> dims beyond 2 unused, no iterate/pad).
// Issued once by the calling wave (EXEC ignored); waits on TENSORcnt.
// ---------------------------------------------------------------------------
__device__ __forceinline__ void tdm_load_tile_2d(const float* gsrc, float* ldsDst,
                                                 int K, int Ncol, int W) {
  unsigned long long ga = (unsigned long long)(uintptr_t)gsrc;
  unsigned lds = (unsigned)(uintptr_t)(void*)ldsDst;  // low 32 bits = LDS offset
  u32x4 g0;
  g0[0] = 1u;                                               // count=1, user mode
  g0[1] = lds;                                              // lds_addr (bytes)
  g0[2] = (unsigned)(ga & 0xFFFFFFFFull);                   // global_addr[31:0]
  g0[3] = (unsigned)((ga >> 32) & 0x01FFFFFFull) | (2u << 30); // ga[56:32]|type=2
  i32x8 g1;
  g1[0] = (int)(2u << 16);                                  // data_size = 4 bytes
  g1[1] = (int)(((unsigned)Ncol & 0xFFFFu) << 16);          // tensor_dim0 lo16
  g1[2] = (int)((((unsigned)Ncol >> 16) & 0xFFFFu) |
                (((unsigned)K & 0xFFFFu) << 16));           // dim0 hi | dim1 lo
  g1[3] = (int)((((unsigned)K >> 16) & 0xFFFFu) |
                (((unsigned)W & 0xFFFFu) << 16));           // dim1 hi | tile_dim0=W
  g1[4] = (int)((unsigned)K & 0xFFFFu);                     // tile_dim1=K, tile_dim2=0
  g1[5] = Ncol;                                             // tensor_dim0_stride lo32
  g1[6] = 0;
  g1[7] = 0;
  i32x4 gz; gz[0] = 0; gz[1] = 0; gz[2] = 0; gz[3] = 0;
#if defined(__clang_major__) && (__clang_major__ >= 23)
  i32x8 gz8; gz8[0]=0; gz8[1]=0; gz8[2]=0; gz8[3]=0; gz8[4]=0; gz8[5]=0; gz8[6]=0; gz8[7]=0;
  __builtin_amdgcn_tensor_load_to_lds(g0, g1, gz, gz, gz8, 0);
#else
  __builtin_amdgcn_tensor_load_to_lds(g0, g1, gz, gz, 0);
#endif
  __builtin_amdgcn_s_wait_tensorcnt(0);
}
#endif  // HAVE_TDM

// ---------------------------------------------------------------------------
// WMMA f32 GEMM: C[M,Ncol] = A[M,K] * B[K,Ncol], row-major fp32.
// Block = 8 waves. Each block owns a K x (NT*16) B tile (TDM-staged in LDS)
// and 8 consecutive 16-row M tiles; each wave keeps NT v8f accumulators and
// issues NT V_WMMA_F32_16X16X4_F32 per A-fragment load (K step 4).
// Fragment layouts per CDNA5 ISA:
//  A 16x4: lanes 0-15 row M=l16 K{k,k+1}; lanes 16-31 K{k+2,k+3}
//  B 4x16: v0 = row k (lanes 0-15) / k+2 (lanes 16-31); v1 = k+1 / k+3
//  C: VGPR j -> M=j (lanes 0-15) / j+8 (lanes 16-31), N=l16
// ---------------------------------------------------------------------------
template <int NT>
__global__ void gemm_wmma_f32(const float* __restrict__ A,
                              const float* __restrict__ B,
                              float* __restrict__ C,
                              int K, int Ncol, int tilesM) {
  extern __shared__ float ldsB[];                 // K * NT*16 floats
  const int W = NT * 16;                          // B tile width in floats
  int nBlocksN = Ncol / W;
  int tnb     = blockIdx.x % nBlocksN;
  int tmBase  = (blockIdx.x / nBlocksN) * 8;
  int wave    = threadIdx.x >> 5;
  int lane    = threadIdx.x & 31;
  int half    = lane >> 4;
  int l16     = lane & 15;

  // ---- stage B[K x W] tile into LDS ----
#if HAVE_TDM
  if (wave == 0) {
    tdm_load_tile_2d(B + tnb * W, ldsB, K, Ncol, W);
  }
#else
  for (int i = threadIdx.x; i < K * W; i += (int)blockDim.x)
    ldsB[i] = B[(size_t)(i / W) * Ncol + tnb * W + (i % W)];
#endif
  __syncthreads();

  int tm = tmBase + wave;
  if (tm >= tilesM) return;

  const float* Arow = A + (size_t)(tm * 16 + l16) * K;

  v8f acc[NT];
#pragma unroll
  for (int c = 0; c < NT; ++c) acc[c] = (v8f){};

  for (int k = 0; k < K; k += 4) {
    v2f a;
    const float* ap = Arow + k + 2 * half;
    a.x = ap[0];
    a.y = ap[1];
    const float* bp = ldsB + (k + 2 * half) * W + l16;
#pragma unroll
    for (int c = 0; c < NT; ++c) {
      v2f b;
      b.x = bp[c * 16];
      b.y = bp[c * 16 + W];
      acc[c] = __builtin_amdgcn_wmma_f32_16x16x4_f32(
          /*neg_a=*/false, a, /*neg_b=*/false, b,
          /*c_mod=*/(short)0, acc[c], /*reuse_a=*/false, /*reuse_b=*/false);
    }
  }

#pragma unroll
  for (int c = 0; c < NT; ++c) {
    float* Cp = C + (size_t)(tm * 16 + 8 * half) * Ncol + tnb * W + c * 16 + l16;
#pragma unroll
    for (int j = 0; j < 8; ++j) Cp[(size_t)j * Ncol] = acc[c][j];
  }
}

// ---------------------------------------------------------------------------
// Per-(node, head) attention scores: a_src/a_dst = <h[n,head,:], att[head,:]>
// ---------------------------------------------------------------------------
__global__ void att_scores(const float* __restrict__ h,
                           const float* __restrict__ att_s,
                           const float* __restrict__ att_d,
                           float* __restrict__ asrc,
                           float* __restrict__ adst,
                           int n, int heads, int ch) {
  int idx = blockIdx.x * blockDim.x + threadIdx.x;
  if (idx >= n * heads) return;
  int head = idx % heads;
  const float* hp = h + (size_t)idx * ch;  // idx*ch == node*heads*ch + head*ch
  const float* sp = att_s + (size_t)head * ch;
  const float* dp = att_d + (size_t)head * ch;
  float s1 = 0.f, s2 = 0.f;
  for (int c = 0; c < ch; ++c) {
    float v = hp[c];
    s1 += v * sp[c];
    s2 += v * dp[c];
  }
  asrc[idx] = s1;
  adst[idx] = s2;
}

__device__ __forceinline__ float edge_score(const float* __restrict__ asrc,
                                            const float* __restrict__ adst,
                                            int src, int dst, int heads, int h) {
  float s = asrc[src * heads + h] + adst[dst * heads + h];
  return (s > 0.f) ? s : NEG_SLOPE * s;   // leaky_relu(0.2)
}

__device__ __forceinline__ void edge_endpoints(const int* __restrict__ srcArr,
                                               const int* __restrict__ dstArr,
                                               int e, int& src, int& dst) {
  if (e < E_EDGES) { src = srcArr[e]; dst = dstArr[e]; }
  else             { src = e - E_EDGES; dst = src; }   // self loop
}

// Pass 1: segment max (monotone-mapped uint atomicMax)
__global__ void edge_max(const int* __restrict__ srcArr,
                         const int* __restrict__ dstArr,
                         const float* __restrict__ asrc,
                         const float* __restrict__ adst,
                         unsigned* __restrict__ maxbuf, int heads) {
  int idx = blockIdx.x * blockDim.x + threadIdx.x;
  if (idx >= ET * heads) return;
  int e = idx / heads, h = idx - e * heads;
  int src, dst;
  edge_endpoints(srcArr, dstArr, e, src, dst);
  float s = edge_score(asrc, adst, src, dst, heads, h);
  atomicMax(&maxbuf[dst * heads + h], mapf(s));
}

// Pass 2: segment sum of exp(score - max)
__global__ void edge_expsum(const int* __restrict__ srcArr,
                            const int* __restrict__ dstArr,
                            const float* __restrict__ asrc,
                            const float* __restrict__ adst,
                            const unsigned* __restrict__ maxbuf,
                            float* __restrict__ sumbuf, int heads) {
  int idx = blockIdx.x * blockDim.x + threadIdx.x;
  if (idx >= ET * heads) return;
  int e = idx / heads, h = idx - e * heads;
  int src, dst;
  edge_endpoints(srcArr, dstArr, e, src, dst);
  float s = edge_score(asrc, adst, src, dst, heads, h);
  float m = unmapf(maxbuf[dst * heads + h]);
  atomicAdd(&sumbuf[dst * heads + h], __expf(s - m));
}

// Pass 3 (layer 1, 8 heads x 32 ch): one wave32 per edge.
// Lanes 0-7 compute the 8 head weights, broadcast via shfl; lane t handles
// channel j*32+t at iteration j (head j) -> coalesced gather + atomic scatter.
__global__ void edge_agg_l1(const int* __restrict__ srcArr,
                            const int* __restrict__ dstArr,
                            const float* __restrict__ asrc,
                            const float* __restrict__ adst,
                            const unsigned* __restrict__ maxbuf,
                            const float* __restrict__ sumbuf,
                            const float* __restrict__ hsrc,
                            float* __restrict__ outacc) {
  int wid  = (int)((blockIdx.x * blockDim.x + threadIdx.x) >> 5);
  int lane = threadIdx.x & 31;
  if (wid >= ET) return;
  int src, dst;
  edge_endpoints(srcArr, dstArr, wid, src, dst);

  const float* hp = hsrc + (size_t)src * C1;
  __builtin_prefetch(hp + lane, 0, 3);   // warm random-indexed source row (L2 hit)

  float w = 0.f;
  if (lane < H1) {
    float s = edge_score(asrc, adst, src, dst, H1, lane);
    float m = unmapf(maxbuf[dst * H1 + lane]);
    float e = __expf(s - m);
    w = e / (sumbuf[dst * H1 + lane] + EPSF);
  }

  float* op = outacc + (size_t)dst * C1;
#pragma unroll
  for (int j = 0; j < H1; ++j) {
    float wj = __shfl(w, j, 32);
    int ch = j * 32 + lane;
    atomicAdd(&op[ch], hp[ch] * wj);
  }
}

// Pass 3 (layer 2, 1 head x 64 ch): one wave32 per edge, accumulate into d_out.
__global__ void edge_agg_l2(const int* __restrict__ srcArr,
                            const int* __restrict__ dstArr,
                            const float* __restrict__ asrc,
                            const float* __restrict__ adst,
                            const unsigned* __restrict__ maxbuf,
                            const float* __restrict__ sumbuf,
                            const float* __restrict__ hsrc,
                            float* __restrict__ outacc) {
  int wid  = (int)((blockIdx.x * blockDim.x + threadIdx.x) >> 5);
  int lane = threadIdx.x & 31;
  if (wid >= ET) return;
  int src, dst;
  edge_endpoints(srcArr, dstArr, wid, src, dst);

  const float* hp = hsrc + (size_t)src * OUT_CH;
  __builtin_prefetch(hp + lane, 0, 3);

  float w = 0.f;
  if (lane == 0) {
    float s = edge_score(asrc, adst, src, dst, 1, 0);
    float m = unmapf(maxbuf[dst]);
    float e = __expf(s - m);
    w = e / (sumbuf[dst] + EPSF);
  }
  w = __shfl(w, 0, 32);

  float* op = outacc + (size_t)dst * OUT_CH;
  atomicAdd(&op[lane],      hp[lane]      * w);
  atomicAdd(&op[lane + 32], hp[lane + 32] * w);
}

// bias (+ optional ReLU), in place
__global__ void bias_act(float* __restrict__ buf, const float* __restrict__ bias,
                         int n, int ch, int doRelu) {
  int idx = blockIdx.x * blockDim.x + threadIdx.x;
  if (idx >= n * ch) return;
  float v = buf[idx] + bias[idx % ch];
  buf[idx] = doRelu ? fmaxf(v, 0.f) : v;
}

// zero accumulators / init segment-max buffers to mapped(-inf) each call
__global__ void init_bufs(float* __restrict__ out1, float* __restrict__ dout,
                          float* __restrict__ sum1, float* __restrict__ sum2,
                          unsigned* __restrict__ max1, unsigned* __restrict__ max2) {
  int idx = blockIdx.x * blockDim.x + threadIdx.x;
  const unsigned NEG_INF_MAPPED = 0x007FFFFFu;  // mapf(-inf)
  if (idx < N_NODES * C1)     out1[idx] = 0.f;
  if (idx < N_NODES * OUT_CH) dout[idx] = 0.f;
  if (idx < N_NODES * H1)   { sum1[idx] = 0.f; max1[idx] = NEG_INF_MAPPED; }
  if (idx < N_NODES)        { sum2[idx] = 0.f; max2[idx] = NEG_INF_MAPPED; }
}

// ---------------------------------------------------------------------------
extern "C" void kernel_launch(void* const* d_in, const int* in_sizes, int n_in,
                              void* d_out, int out_size, void* d_ws, size_t ws_size,
                              hipStream_t stream) {
  (void)in_sizes; (void)n_in; (void)out_size; (void)ws_size;

  const float* x        = (const float*)d_in[0];
  const int*   ei       = (const int*)  d_in[1];   // [2,E]: row0=src, row1=dst
  const float* W1       = (const float*)d_in[2];
  const float* att_src1 = (const float*)d_in[3];
  const float* att_dst1 = (const float*)d_in[4];
  const float* b1       = (const float*)d_in[5];
  const float* W2       = (const float*)d_in[6];
  const float* att_src2 = (const float*)d_in[7];
  const float* att_dst2 = (const float*)d_in[8];
  const float* b2       = (const float*)d_in[9];
  float*       out      = (float*)d_out;

  const int* srcArr = ei;
  const int* dstArr = ei + E_EDGES;

  // ---- workspace carve-up (~122 MB, fits in the 192 MB L2) ----
  float*    ws    = (float*)d_ws;
  float*    h1    = ws;                                   // N*256
  float*    out1  = h1    + (size_t)N_NODES * C1;         // N*256
  float*    asrc1 = out1  + (size_t)N_NODES * C1;         // N*8
  float*    adst1 = asrc1 + (size_t)N_NODES * H1;         // N*8
  float*    sum1  = adst1 + (size_t)N_NODES * H1;         // N*8
  unsigned* max1  = (unsigned*)(sum1 + (size_t)N_NODES * H1);  // N*8
  float*    h2    = (float*)(max1 + (size_t)N_NODES * H1);     // N*64
  float*    asrc2 = h2    + (size_t)N_NODES * OUT_CH;     // N
  float*    adst2 = asrc2 + N_NODES;                      // N
  float*    sum2  = adst2 + N_NODES;                      // N
  unsigned* max2  = (unsigned*)(sum2 + N_NODES);          // N

  const int T = 256;
  #define NBLK(n) ((int)(((long long)(n) + T - 1) / T))

  init_bufs<<<NBLK(N_NODES * C1), T, 0, stream>>>(out1, out, sum1, sum2, max1, max2);

  const int tilesM = N_NODES / 16;          // 3125
  const int tmBlocks = (tilesM + 7) / 8;    // 391

  // ---------------- layer 1 ----------------
  {
    const int NT = 4;                       // 64-wide B tile: 4 col-blocks
    int nBlocksN = C1 / (NT * 16);          // 4
    size_t smem = (size_t)IN_CH * NT * 16 * sizeof(float);  // 32 KB
    gemm_wmma_f32<NT><<<nBlocksN * tmBlocks, T, smem, stream>>>(x, W1, h1,
                                                                IN_CH, C1, tilesM);
  }
  att_scores<<<NBLK(N_NODES * H1), T, 0, stream>>>(h1, att_src1, att_dst1,
                                                   asrc1, adst1, N_NODES, H1, HID);
  edge_max   <<<NBLK(ET * H1), T, 0, stream>>>(srcArr, dstArr, asrc1, adst1, max1, H1);
  edge_expsum<<<NBLK(ET * H1), T, 0, stream>>>(srcArr, dstArr, asrc1, adst1, max1, sum1, H1);
  edge_agg_l1<<<NBLK((long long)ET * 32), T, 0, stream>>>(srcArr, dstArr, asrc1, adst1,
                                                          max1, sum1, h1, out1);
  bias_act<<<NBLK(N_NODES * C1), T, 0, stream>>>(out1, b1, N_NODES, C1, /*relu=*/1);

  // ---------------- layer 2 ----------------
  {
    const int NT = 2;                       // 32-wide B tile: 2 col-blocks
    int nBlocksN = OUT_CH / (NT * 16);      // 2
    size_t smem = (size_t)C1 * NT * 16 * sizeof(float);     // 32 KB
    gemm_wmma_f32<NT><<<nBlocksN * tmBlocks, T, smem, stream>>>(out1, W2, h2,
                                                                C1, OUT_CH, tilesM);
  }
  att_scores<<<NBLK(N_NODES), T, 0, stream>>>(h2, att_src2, att_dst2,
                                              asrc2, adst2, N_NODES, 1, OUT_CH);
  edge_max   <<<NBLK(ET), T, 0, stream>>>(srcArr, dstArr, asrc2, adst2, max2, 1);
  edge_expsum<<<NBLK(ET), T, 0, stream>>>(srcArr, dstArr, asrc2, adst2, max2, sum2, 1);
  edge_agg_l2<<<NBLK((long long)ET * 32), T, 0, stream>>>(srcArr, dstArr, asrc2, adst2,
                                                          max2, sum2, h2, out);
  bias_act<<<NBLK(N_NODES * OUT_CH), T, 0, stream>>>(out, b2, N_NODES, OUT_CH, /*relu=*/0);

  #undef NBLK
}